// Recommender_8796093022752
// MI455X (gfx1250) — compile-verified
//
#include <hip/hip_runtime.h>
#include <hip/hip_bf16.h>

typedef float v2f __attribute__((ext_vector_type(2)));
typedef float v8f __attribute__((ext_vector_type(8)));

// ---------------------------------------------------------------- zero fill
__global__ __launch_bounds__(256) void k_zero(float* __restrict__ p, size_t n) {
  size_t i = (size_t)blockIdx.x * blockDim.x + threadIdx.x;
  size_t stride = (size_t)gridDim.x * blockDim.x;
  for (; i < n; i += stride) p[i] = 0.0f;
}

// ------------------------------------------------- pass 1: per-edge att + seg max
// one wave (32 lanes) per edge; each lane holds a float2 slice of the 64-wide row
__global__ __launch_bounds__(256) void k_edge_att(
    const float* __restrict__ cat, const int* __restrict__ head,
    const int* __restrict__ tail, const int* __restrict__ etype,
    const float* __restrict__ weight, float* __restrict__ att,
    unsigned int* __restrict__ segmax, int E) {
  int wid = (int)((blockIdx.x * (size_t)blockDim.x + threadIdx.x) >> 5);
  if (wid >= E) return;
  int lane = threadIdx.x & 31;
  int h = head[wid], t = tail[wid], r = etype[wid] - 1;
  const float2* hp = (const float2*)(cat + (size_t)h * 64);
  const float2* tp = (const float2*)(cat + (size_t)t * 64);
  const float2* rp = (const float2*)(weight + (size_t)r * 64);
  float2 hv = hp[lane], tv = tp[lane], rv = rp[lane];
  float hx = hv.x * rv.x, hy = hv.y * rv.y;
  float tx = tv.x * rv.x, ty = tv.y * rv.y;
  float hn2 = hx * hx + hy * hy;
  float tn2 = tx * tx + ty * ty;
#pragma unroll
  for (int m = 16; m >= 1; m >>= 1) {
    hn2 += __shfl_xor(hn2, m, 32);
    tn2 += __shfl_xor(tn2, m, 32);
  }
  if (lane == 0) {
    float a = hn2 * tn2;  // (||h*r|| * ||t*r||)^2, always >= 0
    att[wid] = a;
    atomicMax(segmax + h, __float_as_uint(a));  // monotone for non-neg floats
  }
}

// ------------------------------------------------- pass 2: exp + segment sum
__global__ __launch_bounds__(256) void k_expsum(
    const int* __restrict__ head, const float* __restrict__ att,
    const unsigned int* __restrict__ segmax, float* __restrict__ segsum, int E) {
  int e = (int)(blockIdx.x * (size_t)blockDim.x + threadIdx.x);
  if (e >= E) return;
  int h = head[e];
  float mx = __uint_as_float(segmax[h]);
  float v = __expf(att[e] - mx);
  unsafeAtomicAdd(segsum + h, v);  // global_atomic_add_f32
}

// ------------------------------------------------- pass 3: weighted neighbor scatter
__global__ __launch_bounds__(256) void k_edge_agg(
    const float* __restrict__ cat, const int* __restrict__ head,
    const int* __restrict__ tail, const int* __restrict__ etype,
    const float* __restrict__ weight, const float* __restrict__ att,
    const unsigned int* __restrict__ segmax, const float* __restrict__ segsum,
    float* __restrict__ out_cat, int E) {
  int wid = (int)((blockIdx.x * (size_t)blockDim.x + threadIdx.x) >> 5);
  if (wid >= E) return;
  int lane = threadIdx.x & 31;
  int h = head[wid], t = tail[wid], r = etype[wid] - 1;
  float mx = __uint_as_float(segmax[h]);
  float w = __expf(att[wid] - mx) / segsum[h];
  const float2* tp = (const float2*)(cat + (size_t)t * 64);
  const float2* rp = (const float2*)(weight + (size_t)r * 64);
  float2 tv = tp[lane], rv = rp[lane];
  float* dst = out_cat + (size_t)h * 64 + lane * 2;
  unsafeAtomicAdd(dst + 0, w * tv.x * rv.x);
  unsafeAtomicAdd(dst + 1, w * tv.y * rv.y);
}

// ------------------------------------------------- pass 4: interact spmm scatter
__global__ __launch_bounds__(256) void k_interact(
    const float* __restrict__ cat, const int* __restrict__ iu,
    const int* __restrict__ ic, const float* __restrict__ ival,
    float* __restrict__ out_usr, int NNZ) {
  int wid = (int)((blockIdx.x * (size_t)blockDim.x + threadIdx.x) >> 5);
  if (wid >= NNZ) return;
  int lane = threadIdx.x & 31;
  int u = iu[wid], c = ic[wid];
  float v = ival[wid];
  const float2* cp = (const float2*)(cat + (size_t)c * 64);
  float2 cv = cp[lane];
  float* dst = out_usr + (size_t)u * 64 + lane * 2;
  unsafeAtomicAdd(dst + 0, v * cv.x);
  unsafeAtomicAdd(dst + 1, v * cv.y);
}

// ------------------------------------------------- pass 5: WMMA relation gating
// one wave handles a 16-user tile.
// Stage 1: L = U(16x64) @ W^T(64x16)  via V_WMMA_F32_16X16X4_F32, K-steps of 4
// Stage 2: row softmax over 15 relations (half-wave reductions on C layout)
// Stage 3: G = S(16x16) @ W(16x64)    via WMMA (4 N-chunks), then agg *= 1+G
// Weight (15x64) is staged zero-padded into LDS (both orientations) so all B
// fragments are unconditional ds_loads -- no exec-masked padding branches.
__global__ __launch_bounds__(256) void k_user_gate(
    const float* __restrict__ user_emb, const float* __restrict__ weight,
    float* __restrict__ user_agg, int n_users, int n_rel) {
  __shared__ float sW[16][64];     // W, row 15 zeroed          (4 KB)
  __shared__ float sWt[64][16];    // W^T, col 15 zeroed        (4 KB)
  __shared__ float sS[8][16][17];  // per-wave 16x16 score tile (8.7 KB)

  // stage padded weight into LDS (1024 entries / 256 threads)
  for (int i = threadIdx.x; i < 16 * 64; i += 256) {
    int rr = i >> 6, cc = i & 63;
    float v = (rr < n_rel) ? weight[(size_t)rr * 64 + cc] : 0.0f;
    sW[rr][cc] = v;
    sWt[cc][rr] = v;
  }

  int wv = threadIdx.x >> 5;
  int lane = threadIdx.x & 31;
  int l16 = lane & 15;
  int half = lane >> 4;            // 0: lanes 0-15, 1: lanes 16-31
  int koff = half * 2;             // A/B fragment K offset per half-wave
  int base0 = ((int)blockIdx.x * 8 + wv) * 16;
  bool active = (base0 < n_users);
  int base = active ? base0 : 0;   // inactive waves compute on tile 0, never store

  __syncthreads();

  // ---- Stage 1: logits (C layout: lane<16 -> [M=j][N=lane], lane>=16 -> [M=j+8][N=lane-16])
  const float* urow = user_emb + (size_t)(base + l16) * 64;  // A: M = l16 for both halves
  v8f acc = {};
  for (int k0 = 0; k0 < 64; k0 += 4) {
    v2f a, b;
    a.x = urow[k0 + koff];
    a.y = urow[k0 + koff + 1];
    b.x = sWt[k0 + koff][l16];      // B[k][n] = W^T[k][n], col 15 already 0
    b.y = sWt[k0 + koff + 1][l16];
    acc = __builtin_amdgcn_wmma_f32_16x16x4_f32(false, a, false, b, (short)0,
                                                acc, false, false);
  }

  // ---- Stage 2: softmax across relations (row = VGPR slot, spread over a 16-lane group)
  bool nvalid = (l16 < n_rel);
  v8f sc;
#pragma unroll
  for (int j = 0; j < 8; ++j) {
    float x = nvalid ? acc[j] : -3.0e38f;
#pragma unroll
    for (int m = 8; m >= 1; m >>= 1) x = fmaxf(x, __shfl_xor(x, m, 32));
    float ev = nvalid ? __expf(acc[j] - x) : 0.0f;
    float sm = ev;
#pragma unroll
    for (int m = 8; m >= 1; m >>= 1) sm += __shfl_xor(sm, m, 32);
    sc[j] = ev / sm;
  }

  // ---- C-layout -> LDS so we can re-read S as A-layout fragments
#pragma unroll
  for (int j = 0; j < 8; ++j) sS[wv][j + half * 8][l16] = sc[j];
  __syncthreads();

  // ---- Stage 3: G = S @ W over 4 column chunks of 16
  for (int c = 0; c < 4; ++c) {
    v8f g = {};
#pragma unroll
    for (int k0 = 0; k0 < 16; k0 += 4) {
      int k = k0 + koff;
      int col = c * 16 + l16;
      v2f a, b;
      a.x = sS[wv][l16][k];         // A: M = l16, K = k
      a.y = sS[wv][l16][k + 1];
      b.x = sW[k][col];             // row 15 already zeroed
      b.y = sW[k + 1][col];
      g = __builtin_amdgcn_wmma_f32_16x16x4_f32(false, a, false, b, (short)0,
                                                g, false, false);
    }
    if (active) {
#pragma unroll
      for (int j = 0; j < 8; ++j) {  // C layout: row = j + half*8, col = c*16 + l16
        size_t idx = (size_t)(base + j + half * 8) * 64 + c * 16 + l16;
        user_agg[idx] = user_agg[idx] * (1.0f + g[j]);
      }
    }
  }
}

// ---------------------------------------------------------------- launcher
extern "C" void kernel_launch(void* const* d_in, const int* in_sizes, int n_in,
                              void* d_out, int out_size, void* d_ws, size_t ws_size,
                              hipStream_t stream) {
  const float* cat    = (const float*)d_in[0];
  const float* usr    = (const float*)d_in[1];
  const int*   eidx   = (const int*)d_in[2];
  const int*   etype  = (const int*)d_in[3];
  const int*   iidx   = (const int*)d_in[4];
  const float* ival   = (const float*)d_in[5];
  const float* weight = (const float*)d_in[6];

  int ncat   = in_sizes[0] / 64;
  int nusers = in_sizes[1] / 64;
  int E      = in_sizes[2] / 2;
  int nnz    = in_sizes[5];
  int nrel   = in_sizes[6] / 64;  // 15

  const int* head = eidx;
  const int* tail = eidx + E;
  const int* iu   = iidx;
  const int* ic   = iidx + nnz;

  float* out_cat = (float*)d_out;
  float* out_usr = out_cat + (size_t)ncat * 64;

  float*        att    = (float*)d_ws;              // E floats
  unsigned int* segmax = (unsigned int*)(att + E);  // ncat u32 (bits of f32 >= 0)
  float*        segsum = (float*)(segmax + ncat);   // ncat floats

  k_zero<<<2048, 256, 0, stream>>>((float*)d_out, (size_t)out_size);
  k_zero<<<256, 256, 0, stream>>>((float*)segmax, (size_t)ncat * 2);  // segmax+segsum

  int eblocks = (E + 7) / 8;      // 8 waves (edges) per 256-thread block
  k_edge_att<<<eblocks, 256, 0, stream>>>(cat, head, tail, etype, weight, att, segmax, E);
  k_expsum<<<(E + 255) / 256, 256, 0, stream>>>(head, att, segmax, segsum, E);
  k_edge_agg<<<eblocks, 256, 0, stream>>>(cat, head, tail, etype, weight, att,
                                          segmax, segsum, out_cat, E);

  int iblocks = (nnz + 7) / 8;
  k_interact<<<iblocks, 256, 0, stream>>>(cat, iu, ic, ival, out_usr, nnz);

  int utiles = (nusers + 15) / 16;
  k_user_gate<<<(utiles + 7) / 8, 256, 0, stream>>>(usr, weight, out_usr, nusers, nrel);
}